// TemporalGraphNetwork_43619687858844
// MI455X (gfx1250) — compile-verified
//
#include <hip/hip_runtime.h>
#include <math.h>
#include <stdint.h>

// ---------------- problem constants ----------------
#define NNODE 20000
#define NEDGE 320000
#define NNEVT 2000
#define NEEVT 2000
#define ETE   16
#define MEMD  64
#define TENC  32
#define EEMB  128
#define OUTD  64
#define HEADS 4
#define HD    256      // HEADS*OUTD
#define MSGD  304      // ETE + 2*MEM + TENC + EEMB
#define EDIM  160      // TENC + EEMB
#define GDIM  192      // 3*MEM (GRU gates)

typedef __attribute__((ext_vector_type(2))) float v2f;
typedef __attribute__((ext_vector_type(8))) float v8f;
typedef __attribute__((ext_vector_type(4))) unsigned int u32x4;
typedef __attribute__((ext_vector_type(8))) int i32x8;
typedef __attribute__((ext_vector_type(4))) int i32x4;

static __device__ __forceinline__ float sigmoidf_(float x) { return 1.f / (1.f + expf(-x)); }

static __device__ __forceinline__ void atomicMaxFloat(float* addr, float val) {
    // IEEE monotonic bit trick: ints compare like floats for >=0; uints reversed for <0.
    if (val >= 0.f) atomicMax((int*)addr, __float_as_int(val));
    else            atomicMin((unsigned int*)addr, __float_as_uint(val));
}

// ---------------- TDM: DMA a (rows x 16 f32) weight panel into LDS ----------------
// Descriptor per CDNA5 ISA 8.3/8.4 (D# groups 0/1; groups 2/3 zero -> 2D).
// LDS padding: 1 dword after every 16 dwords -> LDS row stride 17 floats
// (pad_interval code 3 = 16 dwords, pad_amount code 0 = 1 dword).
// Toolchain uses the 6-arg builtin: (u32x4, i32x8, i32x4, i32x4, i32x8, i32 cpol).
static __device__ __forceinline__ void tdm_load_B_panel(
    const float* gsrc, int ld, int rows, float* ldsDst)
{
    unsigned long long ga = (unsigned long long)(uintptr_t)gsrc;
    unsigned lds = (unsigned)(unsigned long long)(uintptr_t)ldsDst; // LDS byte offset
    u32x4 g0 = { 1u,                                   // count = 1 valid descriptor
                 lds,                                  // lds_addr
                 (unsigned)ga,                         // global_addr[31:0]
                 (unsigned)((ga >> 32) & 0x1ffffffu)   // global_addr[56:32]
                   | 0x80000000u };                    // type = 2 ("image")
    i32x8 g1 = { (2 << 16)                             // data_size = 4 bytes
               | (1 << 20)                             // pad_enable
               | (3 << 22)                             // pad_interval: 16 dwords
               | (0 << 25),                            // pad_amount: 1 dword
                 (16 << 16),                           // tensor_dim0 = 16 (bits 79:48 lo)
                 ((rows & 0xffff) << 16),              // tensor_dim1 = rows (bits 111:80 lo)
                 (16 << 16),                           // tile_dim0 = 16 (bits 127:112)
                 (rows & 0xffff),                      // tile_dim1 = rows; tile_dim2 = 0
                 ld,                                   // tensor_dim0_stride (bits 207:160 lo)
                 0, 0 };
    i32x4 gz4 = { 0, 0, 0, 0 };
    i32x8 gz8 = { 0, 0, 0, 0, 0, 0, 0, 0 };
    __builtin_amdgcn_tensor_load_to_lds(g0, g1, gz4, gz4, gz8, 0);
}

// ---------------- generic fp32 WMMA GEMM ----------------
// C[M,N] = A[M,K] @ B[K,N] (+bias).  Block: 128 thr = 4 waves; tile 64(M)x16(N).
// B panel staged via TDM (wave 0), A tile via cooperative vector loads.
// Fragment layouts per CDNA5 ISA 7.12.2:
//   A 16x4 : lane l, elem r -> m=l&15, k=r+2*(l>>4)
//   B 4x16 : lane l, elem r -> n=l&15, k=r+2*(l>>4)
//   C 16x16: lane l, elem r -> m=r+8*(l>>4), n=l&15
#define KC 32

__global__ __launch_bounds__(128) void k_gemm(
    const float* __restrict__ A, const float* __restrict__ B,
    const float* __restrict__ bias, float* __restrict__ C,
    int M, int N, int K, int lda, int ldb, int ldc)
{
    __shared__ __align__(16) float sA[64 * (KC + 1)];
    __shared__ __align__(16) float sB[KC * 17];
    const int tid  = threadIdx.x;
    const int wave = tid >> 5, lane = tid & 31;
    const int hi   = lane >> 4, lo = lane & 15;
    const int rowBlk = blockIdx.y * 64;
    const int colBlk = blockIdx.x * 16;

    v8f acc = {0.f, 0.f, 0.f, 0.f, 0.f, 0.f, 0.f, 0.f};

    for (int k0 = 0; k0 < K; k0 += KC) {
        const int kc = min(KC, K - k0);
        if (wave == 0)   // async DMA of the B panel while A is loaded below
            tdm_load_B_panel(B + (long)k0 * ldb + colBlk, ldb, kc, sB);
        for (int idx = tid; idx < 64 * kc; idx += 128) {
            int r = idx / kc, c = idx - r * kc;
            int gr = rowBlk + r;
            sA[r * (KC + 1) + c] = (gr < M) ? A[(long)gr * lda + k0 + c] : 0.f;
        }
        if (wave == 0) __builtin_amdgcn_s_wait_tensorcnt(0);
        __syncthreads();
        const float* aRow = sA + (wave * 16 + lo) * (KC + 1);
        if (kc == KC) {
            v2f af[8], bf[8];
            #pragma unroll
            for (int t = 0; t < 8; ++t) {
                const int kk = t * 4;
                af[t].x = aRow[kk + 2 * hi];
                af[t].y = aRow[kk + 2 * hi + 1];
                bf[t].x = sB[(kk + 2 * hi) * 17 + lo];
                bf[t].y = sB[(kk + 2 * hi + 1) * 17 + lo];
            }
            #pragma unroll
            for (int t = 0; t < 8; ++t)
                acc = __builtin_amdgcn_wmma_f32_16x16x4_f32(false, af[t], false, bf[t],
                                                            (short)0, acc, false, false);
        } else {
            for (int kk = 0; kk < kc; kk += 4) {
                v2f a, b;
                a.x = aRow[kk + 2 * hi];
                a.y = aRow[kk + 2 * hi + 1];
                b.x = sB[(kk + 2 * hi) * 17 + lo];
                b.y = sB[(kk + 2 * hi + 1) * 17 + lo];
                acc = __builtin_amdgcn_wmma_f32_16x16x4_f32(false, a, false, b,
                                                            (short)0, acc, false, false);
            }
        }
        __syncthreads();
    }
    const float bv = bias ? bias[colBlk + lo] : 0.f;
    #pragma unroll
    for (int r = 0; r < 8; ++r) {
        int gr = rowBlk + wave * 16 + r + 8 * hi;
        if (gr < M) C[(long)gr * ldc + colBlk + lo] = acc[r] + bv;
    }
}

// ---------------- fused edge-attr projection GEMM ----------------
// E[NEDGE,256] = [cos((ts-lu)*tw+tb) || edge_features] @ We[160,256]
// A is synthesized in the LDS loader (edge_attr never materialized).
__global__ __launch_bounds__(128) void k_egemm(
    const float* __restrict__ ets, const float* __restrict__ elu,
    const float* __restrict__ tw,  const float* __restrict__ tb,
    const float* __restrict__ efeat, const float* __restrict__ We,
    float* __restrict__ E)
{
    __shared__ __align__(16) float sA[64 * (KC + 1)];
    __shared__ __align__(16) float sB[KC * 17];
    const int tid  = threadIdx.x;
    const int wave = tid >> 5, lane = tid & 31;
    const int hi   = lane >> 4, lo = lane & 15;
    const int rowBlk = blockIdx.y * 64;      // edges, NEDGE % 64 == 0
    const int colBlk = blockIdx.x * 16;

    v8f acc = {0.f, 0.f, 0.f, 0.f, 0.f, 0.f, 0.f, 0.f};

    for (int k0 = 0; k0 < EDIM; k0 += KC) {  // 160 = 5 * 32
        if (wave == 0)
            tdm_load_B_panel(We + (long)k0 * HD + colBlk, HD, KC, sB);
        for (int idx = tid; idx < 64 * KC; idx += 128) {
            int r = idx >> 5, c = idx & 31;   // KC == 32
            int gr = rowBlk + r;
            int cc = k0 + c;
            float v;
            if (cc < TENC) v = cosf((ets[gr] - elu[gr]) * tw[cc] + tb[cc]);
            else           v = efeat[(long)gr * EEMB + (cc - TENC)];
            sA[r * (KC + 1) + c] = v;
        }
        if (wave == 0) __builtin_amdgcn_s_wait_tensorcnt(0);
        __syncthreads();
        const float* aRow = sA + (wave * 16 + lo) * (KC + 1);
        v2f af[8], bf[8];
        #pragma unroll
        for (int t = 0; t < 8; ++t) {
            const int kk = t * 4;
            af[t].x = aRow[kk + 2 * hi];
            af[t].y = aRow[kk + 2 * hi + 1];
            bf[t].x = sB[(kk + 2 * hi) * 17 + lo];
            bf[t].y = sB[(kk + 2 * hi + 1) * 17 + lo];
        }
        #pragma unroll
        for (int t = 0; t < 8; ++t)
            acc = __builtin_amdgcn_wmma_f32_16x16x4_f32(false, af[t], false, bf[t],
                                                        (short)0, acc, false, false);
        __syncthreads();
    }
    #pragma unroll
    for (int r = 0; r < 8; ++r) {
        int gr = rowBlk + wave * 16 + r + 8 * hi;
        E[(long)gr * HD + colBlk + lo] = acc[r];
    }
}

// ---------------- small utility kernels ----------------
__global__ void k_fill(float* __restrict__ p, float v, long n) {
    long i = (long)blockIdx.x * blockDim.x + threadIdx.x;
    if (i < n) p[i] = v;
}

__global__ void k_transpose(const float* __restrict__ in, float* __restrict__ out,
                            int R, int Ccols) {
    long i = (long)blockIdx.x * blockDim.x + threadIdx.x;
    if (i >= (long)R * Ccols) return;
    int r = (int)(i / Ccols), c = (int)(i - (long)r * Ccols);
    out[(long)c * R + r] = in[i];
}

__global__ void k_concat2(const float* __restrict__ a, int wa,
                          const float* __restrict__ b, int wb,
                          float* __restrict__ o, int rows) {
    const int w = wa + wb;
    long i = (long)blockIdx.x * blockDim.x + threadIdx.x;
    if (i >= (long)rows * w) return;
    int r = (int)(i / w), c = (int)(i - (long)r * w);
    o[i] = (c < wa) ? a[(long)r * wa + c] : b[(long)r * wb + (c - wa)];
}

// one wave per message (node evt, edge-src msg, edge-dst msg); atomic scatter into agg
__global__ void k_messages(
    const int* __restrict__ ne_type, const int* __restrict__ ne_node,
    const float* __restrict__ ne_emb, const float* __restrict__ ne_ts,
    const float* __restrict__ ne_mask,
    const int* __restrict__ ee_type, const int* __restrict__ ee_src,
    const int* __restrict__ ee_dst, const int* __restrict__ ee_eid,
    const float* __restrict__ ee_emb, const float* __restrict__ ee_ts,
    const float* __restrict__ ee_mask,
    const float* __restrict__ memory, const float* __restrict__ elu,
    const float* __restrict__ evt_emb, const float* __restrict__ tw,
    const float* __restrict__ tb,
    float* __restrict__ agg, int* __restrict__ touched)
{
    int wid  = (int)(((long)blockIdx.x * blockDim.x + threadIdx.x) >> 5);
    int lane = threadIdx.x & 31;
    if (wid >= NNEVT + 2 * NEEVT) return;

    int node; float mask, t;
    const float* ete; const float* emb; const float* memA; const float* memB;
    if (wid < NNEVT) {
        int ev = wid;
        node = ne_node[ev]; mask = ne_mask[ev]; t = ne_ts[ev];
        ete  = evt_emb + ne_type[ev] * ETE;
        emb  = ne_emb + (long)ev * EEMB;
        memA = memory + (long)node * MEMD;
        memB = nullptr;                       // zeros segment
    } else {
        int kind = (wid < NNEVT + NEEVT) ? 1 : 2;  // 1 = src msg, 2 = dst msg
        int ev   = (kind == 1) ? wid - NNEVT : wid - NNEVT - NEEVT;
        int s = ee_src[ev], d = ee_dst[ev];
        node = (kind == 1) ? s : d;
        mask = ee_mask[ev];
        t    = ee_ts[ev] - elu[ee_eid[ev]];
        ete  = evt_emb + ee_type[ev] * ETE;
        emb  = ee_emb + (long)ev * EEMB;
        memA = memory + (long)((kind == 1) ? s : d) * MEMD;
        memB = memory + (long)((kind == 1) ? d : s) * MEMD;
    }
    float* dstp = agg + (long)node * MSGD;
    for (int j = lane; j < MSGD; j += 32) {
        float v;
        if      (j < ETE)            v = ete[j];
        else if (j < ETE + MEMD)     v = memA[j - ETE];
        else if (j < ETE + 2 * MEMD) v = memB ? memB[j - ETE - MEMD] : 0.f;
        else if (j < ETE + 2 * MEMD + TENC) {
            int c = j - ETE - 2 * MEMD;
            v = cosf(t * tw[c] + tb[c]);
        } else                       v = emb[j - ETE - 2 * MEMD - TENC];
        atomicAdd(dstp + j, v * mask);
    }
    if (lane == 0) atomicAdd(touched + node, 1);
}

__global__ void k_gru(const float* __restrict__ gi, const float* __restrict__ gh,
                      const float* __restrict__ mem, const int* __restrict__ touched,
                      float* __restrict__ nm) {
    long i = (long)blockIdx.x * blockDim.x + threadIdx.x;
    if (i >= (long)NNODE * MEMD) return;
    int n = (int)(i >> 6), j = (int)(i & 63);
    const float* a = gi + (long)n * GDIM;
    const float* b = gh + (long)n * GDIM;
    float r  = sigmoidf_(a[j] + b[j]);
    float z  = sigmoidf_(a[MEMD + j] + b[MEMD + j]);
    float nn = tanhf(a[2 * MEMD + j] + r * b[2 * MEMD + j]);
    float h  = mem[i];
    float v  = (1.f - z) * nn + z * h;
    nm[i] = (touched[n] > 0) ? v : h;
}

__global__ void k_alpha(const float* __restrict__ q, const float* __restrict__ k,
                        const float* __restrict__ ep, const int* __restrict__ src,
                        const int* __restrict__ dst, float* __restrict__ alpha,
                        float* __restrict__ mbuf) {
    long i = (long)blockIdx.x * blockDim.x + threadIdx.x;
    if (i >= (long)NEDGE * HEADS) return;
    int e = (int)(i >> 2), h = (int)(i & 3);
    int s = src[e], d = dst[e];
    const float* qp = q  + (long)d * HD + h * OUTD;
    const float* kp = k  + (long)s * HD + h * OUTD;
    const float* cp = ep + (long)e * HD + h * OUTD;
    float a = 0.f;
    #pragma unroll 8
    for (int j = 0; j < OUTD; ++j) a += qp[j] * (kp[j] + cp[j]);
    a *= 0.125f;                       // 1/sqrt(64)
    alpha[i] = a;
    atomicMaxFloat(mbuf + (long)d * HEADS + h, a);
}

__global__ void k_expden(float* __restrict__ alpha, const float* __restrict__ mbuf,
                         const int* __restrict__ dst, float* __restrict__ den) {
    long i = (long)blockIdx.x * blockDim.x + threadIdx.x;
    if (i >= (long)NEDGE * HEADS) return;
    int e = (int)(i >> 2), h = (int)(i & 3);
    float m = mbuf[(long)dst[e] * HEADS + h];
    if (!isfinite(m)) m = 0.f;
    float ex = expf(alpha[i] - m);
    alpha[i] = ex;                     // overwrite with exp
    atomicAdd(den + (long)dst[e] * HEADS + h, ex);
}

// one wave per edge: out[dst] += (v[src] + e) * softmax_w  (on top of skip GEMM result)
__global__ void k_aggregate(const float* __restrict__ v, const float* __restrict__ ep,
                            const float* __restrict__ ex, const float* __restrict__ den,
                            const int* __restrict__ src, const int* __restrict__ dstv,
                            float* __restrict__ out) {
    int wid  = (int)(((long)blockIdx.x * blockDim.x + threadIdx.x) >> 5);
    int lane = threadIdx.x & 31;
    if (wid >= NEDGE) return;
    int s = src[wid], d = dstv[wid];
    float w0 = ex[(long)wid * 4 + 0] / (den[(long)d * 4 + 0] + 1e-16f);
    float w1 = ex[(long)wid * 4 + 1] / (den[(long)d * 4 + 1] + 1e-16f);
    float w2 = ex[(long)wid * 4 + 2] / (den[(long)d * 4 + 2] + 1e-16f);
    float w3 = ex[(long)wid * 4 + 3] / (den[(long)d * 4 + 3] + 1e-16f);
    for (int j = lane; j < HD; j += 32) {
        float w = (j < 64) ? w0 : (j < 128) ? w1 : (j < 192) ? w2 : w3;
        atomicAdd(out + (long)d * HD + j,
                  (v[(long)s * HD + j] + ep[(long)wid * HD + j]) * w);
    }
}

// ---------------- host orchestration ----------------
static inline void fill_f(float* p, float v, long n, hipStream_t s) {
    k_fill<<<dim3((unsigned)((n + 255) / 256)), dim3(256), 0, s>>>(p, v, n);
}

extern "C" void kernel_launch(void* const* d_in, const int* in_sizes, int n_in,
                              void* d_out, int out_size, void* d_ws, size_t ws_size,
                              hipStream_t stream) {
    // ---- inputs (setup_inputs order) ----
    const int*   ne_type = (const int*)  d_in[0];
    const int*   ne_node = (const int*)  d_in[1];
    const float* ne_emb  = (const float*)d_in[2];
    const float* ne_ts   = (const float*)d_in[3];
    const float* ne_mask = (const float*)d_in[4];
    const int*   ee_type = (const int*)  d_in[5];
    const int*   ee_src  = (const int*)  d_in[6];
    const int*   ee_dst  = (const int*)  d_in[7];
    const int*   ee_eid  = (const int*)  d_in[8];
    const float* ee_emb  = (const float*)d_in[9];
    const float* ee_ts   = (const float*)d_in[10];
    const float* ee_mask = (const float*)d_in[11];
    const float* memory  = (const float*)d_in[12];
    const float* nfeat   = (const float*)d_in[13];
    const int*   eidx    = (const int*)  d_in[14];
    const float* efeat   = (const float*)d_in[15];
    const float* ets     = (const float*)d_in[16];
    const float* elu     = (const float*)d_in[17];
    // ---- params: jax pytree (sorted-key DFS) flattening assumption ----
    // Wl, bl, convs[0]{We,Wk,Wq,Wskip,Wv,bk,bq,bskip,bv}, convs[1]{...},
    // evt_emb, gru{W_hh,W_ih,b_hh,b_ih}, tb, tw
    const float* Wl      = (const float*)d_in[18];
    const float* bl      = (const float*)d_in[19];
    const float* We[2]    = {(const float*)d_in[20], (const float*)d_in[29]};
    const float* Wk[2]    = {(const float*)d_in[21], (const float*)d_in[30]};
    const float* Wq[2]    = {(const float*)d_in[22], (const float*)d_in[31]};
    const float* Wskip[2] = {(const float*)d_in[23], (const float*)d_in[32]};
    const float* Wv[2]    = {(const float*)d_in[24], (const float*)d_in[33]};
    const float* bk[2]    = {(const float*)d_in[25], (const float*)d_in[34]};
    const float* bq[2]    = {(const float*)d_in[26], (const float*)d_in[35]};
    const float* bskip[2] = {(const float*)d_in[27], (const float*)d_in[36]};
    const float* bv[2]    = {(const float*)d_in[28], (const float*)d_in[37]};
    const float* evt_emb = (const float*)d_in[38];
    const float* W_hh    = (const float*)d_in[39];  // [192,64]
    const float* W_ih    = (const float*)d_in[40];  // [192,304]
    const float* b_hh    = (const float*)d_in[41];
    const float* b_ih    = (const float*)d_in[42];
    const float* tb      = (const float*)d_in[43];
    const float* tw      = (const float*)d_in[44];
    const int* src = eidx;
    const int* dst = eidx + NEDGE;

    // ---- workspace carve-up ----
    char* base = (char*)d_ws;
    size_t off = 0;
    auto alloc = [&](size_t bytes) -> char* {
        off = (off + 255) & ~(size_t)255;
        char* p = base + off;
        off += bytes;
        return p;
    };
    float* fWihT = (float*)alloc((size_t)MSGD * GDIM * 4);   // [304,192]
    float* fWhhT = (float*)alloc((size_t)MEMD * GDIM * 4);   // [64,192]
    float* fAgg  = (float*)alloc((size_t)NNODE * MSGD * 4);
    int*   iTch  = (int*)  alloc((size_t)NNODE * 4);
    float* fGi   = (float*)alloc((size_t)NNODE * GDIM * 4);
    float* fGh   = (float*)alloc((size_t)NNODE * GDIM * 4);
    float* fNm   = (float*)alloc((size_t)NNODE * MEMD * 4);
    float* fX    = (float*)alloc((size_t)NNODE * (EEMB + MEMD) * 4);      // 192
    float* fX1   = (float*)alloc((size_t)NNODE * (EEMB + MEMD + HD) * 4); // 448
    float* fE    = (float*)alloc((size_t)NEDGE * HD * 4);    // 327.7 MB
    float* fQ    = (float*)alloc((size_t)NNODE * HD * 4);
    float* fK    = (float*)alloc((size_t)NNODE * HD * 4);
    float* fV    = (float*)alloc((size_t)NNODE * HD * 4);
    float* fH    = (float*)alloc((size_t)NNODE * HD * 4);
    float* fAl   = (float*)alloc((size_t)NEDGE * HEADS * 4);
    float* fM    = (float*)alloc((size_t)NNODE * HEADS * 4);
    float* fDen  = (float*)alloc((size_t)NNODE * HEADS * 4);
    (void)ws_size; (void)in_sizes; (void)n_in; (void)out_size;

    const dim3 B256(256);
    // ---- 1. weight transposes (x @ W^T form) ----
    k_transpose<<<dim3((GDIM * MSGD + 255) / 256), B256, 0, stream>>>(W_ih, fWihT, GDIM, MSGD);
    k_transpose<<<dim3((GDIM * MEMD + 255) / 256), B256, 0, stream>>>(W_hh, fWhhT, GDIM, MEMD);

    // ---- 2. message scatter-sum ----
    fill_f(fAgg, 0.f, (long)NNODE * MSGD, stream);
    fill_f((float*)iTch, 0.f, NNODE, stream);  // bit pattern 0 == int 0
    {
        int waves = NNEVT + 2 * NEEVT;
        k_messages<<<dim3((waves * 32 + 127) / 128), dim3(128), 0, stream>>>(
            ne_type, ne_node, ne_emb, ne_ts, ne_mask,
            ee_type, ee_src, ee_dst, ee_eid, ee_emb, ee_ts, ee_mask,
            memory, elu, evt_emb, tw, tb, fAgg, iTch);
    }

    // ---- 3. GRU memory update ----
    k_gemm<<<dim3(GDIM / 16, (NNODE + 63) / 64), dim3(128), 0, stream>>>(
        fAgg, fWihT, b_ih, fGi, NNODE, GDIM, MSGD, MSGD, GDIM, GDIM);
    k_gemm<<<dim3(GDIM / 16, (NNODE + 63) / 64), dim3(128), 0, stream>>>(
        memory, fWhhT, b_hh, fGh, NNODE, GDIM, MEMD, MEMD, GDIM, GDIM);
    k_gru<<<dim3((NNODE * MEMD + 255) / 256), B256, 0, stream>>>(fGi, fGh, memory, iTch, fNm);

    // ---- 4. x = [node_features || memory'] ----
    k_concat2<<<dim3((NNODE * 192 + 255) / 256), B256, 0, stream>>>(nfeat, EEMB, fNm, MEMD, fX, NNODE);

    // ---- 5. two TransformerConv blocks ----
    for (int c = 0; c < 2; ++c) {
        const float* xin = (c == 0) ? fX : fX1;
        const int    kd  = (c == 0) ? 192 : 448;
        // edge projection (fused time-enc + concat + GEMM)
        k_egemm<<<dim3(HD / 16, NEDGE / 64), dim3(128), 0, stream>>>(
            ets, elu, tw, tb, efeat, We[c], fE);
        // node projections
        k_gemm<<<dim3(HD / 16, (NNODE + 63) / 64), dim3(128), 0, stream>>>(
            xin, Wq[c], bq[c], fQ, NNODE, HD, kd, kd, HD, HD);
        k_gemm<<<dim3(HD / 16, (NNODE + 63) / 64), dim3(128), 0, stream>>>(
            xin, Wk[c], bk[c], fK, NNODE, HD, kd, kd, HD, HD);
        k_gemm<<<dim3(HD / 16, (NNODE + 63) / 64), dim3(128), 0, stream>>>(
            xin, Wv[c], bv[c], fV, NNODE, HD, kd, kd, HD, HD);
        k_gemm<<<dim3(HD / 16, (NNODE + 63) / 64), dim3(128), 0, stream>>>(
            xin, Wskip[c], bskip[c], fH, NNODE, HD, kd, kd, HD, HD);  // skip term first
        // segment softmax over dst
        fill_f(fM, -INFINITY, (long)NNODE * HEADS, stream);
        fill_f(fDen, 0.f, (long)NNODE * HEADS, stream);
        k_alpha<<<dim3((NEDGE * HEADS + 255) / 256), B256, 0, stream>>>(fQ, fK, fE, src, dst, fAl, fM);
        k_expden<<<dim3((NEDGE * HEADS + 255) / 256), B256, 0, stream>>>(fAl, fM, dst, fDen);
        // weighted aggregation, atomically added onto the skip result
        k_aggregate<<<dim3((NEDGE * 32 + 255) / 256), B256, 0, stream>>>(fV, fE, fAl, fDen, src, dst, fH);
        if (c == 0) {  // x1 = [h || x]
            k_concat2<<<dim3((NNODE * 448 + 255) / 256), B256, 0, stream>>>(fH, HD, fX, 192, fX1, NNODE);
        }
    }

    // ---- 6. output projection ----
    k_gemm<<<dim3(OUTD / 16, (NNODE + 63) / 64), dim3(128), 0, stream>>>(
        fH, Wl, bl, (float*)d_out, NNODE, OUTD, HD, HD, OUTD, OUTD);
}